// GNNLayer_34213709480007
// MI455X (gfx1250) — compile-verified
//
#include <hip/hip_runtime.h>
#include <math.h>

#define NNODES 50000
#define NEDGES 800000
#define NEG_SLOPE 0.2f

typedef float v2f __attribute__((ext_vector_type(2)));
typedef float v8f __attribute__((ext_vector_type(8)));

// ---------------------------------------------------------------------------
// WMMA f32 GEMM: C[M,Nc] = A[M,K] * B[K,Nc], all row-major.
// Requires M%16==0, Nc%16==0, K%4==0. One wave computes one 16x16 C tile via
// a chain of V_WMMA_F32_16X16X4_F32 (K stepped by 4).
// ---------------------------------------------------------------------------
__global__ void gemm_wmma_f32(const float* __restrict__ A, const float* __restrict__ B,
                              float* __restrict__ C, int M, int K, int Nc) {
    int wave = threadIdx.x >> 5;
    int lane = threadIdx.x & 31;
    int tiles_n = Nc >> 4;
    int total = (M >> 4) * tiles_n;
    int tile = blockIdx.x * 8 + wave;
    if (tile >= total) return;
    int tm = tile / tiles_n;
    int tn = tile - tm * tiles_n;
    int r0 = tm << 4, c0 = tn << 4;
    int hi = lane >> 4;       // 0 or 1
    int lm = lane & 15;

    // A fragment: lane holds A[r0+lm][k + 2*hi + {0,1}]
    const float* Arow = A + (size_t)(r0 + lm) * K + 2 * hi;
    // B fragment: lane holds B[k + 2*hi + {0,1}][c0+lm]
    const float* Bcol = B + (size_t)(2 * hi) * Nc + (c0 + lm);

    v8f acc = {0.f, 0.f, 0.f, 0.f, 0.f, 0.f, 0.f, 0.f};
    for (int k = 0; k < K; k += 4) {
#if __has_builtin(__builtin_amdgcn_wmma_f32_16x16x4_f32)
        v2f a, b;
        a.x = Arow[k];
        a.y = Arow[k + 1];
        b.x = Bcol[(size_t)k * Nc];
        b.y = Bcol[(size_t)(k + 1) * Nc];
        acc = __builtin_amdgcn_wmma_f32_16x16x4_f32(
            /*neg_a=*/false, a, /*neg_b=*/false, b,
            /*c_mod=*/(short)0, acc, /*reuse_a=*/false, /*reuse_b=*/false);
#else
        // scalar fallback (keeps compile green; histogram will show wmma==0)
        #pragma unroll
        for (int j = 0; j < 8; ++j) {
            const float* ar = A + (size_t)(r0 + j + 8 * hi) * K + k;
            const float* bc = B + (size_t)k * Nc + (c0 + lm);
            float s = acc[j];
            s += ar[0] * bc[0];
            s += ar[1] * bc[(size_t)Nc];
            s += ar[2] * bc[(size_t)2 * Nc];
            s += ar[3] * bc[(size_t)3 * Nc];
            acc[j] = s;
        }
#endif
    }
    // C fragment: VGPR j -> row r0 + j + 8*hi, col c0 + lm
    float* Crow = C + (size_t)(r0 + 8 * hi) * Nc + (c0 + lm);
    #pragma unroll
    for (int j = 0; j < 8; ++j) Crow[(size_t)j * Nc] = acc[j];
}

// ---------------------------------------------------------------------------
// Per-node attention logits: out_s[n,h] = sum_c h[n,h,c]*a_s[h,c]  (same for d)
// One block per node, HC threads, LDS tree reduction within each head.
// ---------------------------------------------------------------------------
template <int HC, int C>
__global__ void alpha_kernel(const float* __restrict__ h, const float* __restrict__ a_s,
                             const float* __restrict__ a_d, float* __restrict__ out_s,
                             float* __restrict__ out_d) {
    __shared__ float ss[HC];
    __shared__ float sd[HC];
    int node = blockIdx.x;
    int t = threadIdx.x;
    float v = h[(size_t)node * HC + t];
    ss[t] = v * a_s[t];
    sd[t] = v * a_d[t];
    __syncthreads();
    #pragma unroll
    for (int s = C >> 1; s > 0; s >>= 1) {
        if ((t & (C - 1)) < s) {
            ss[t] += ss[t + s];
            sd[t] += sd[t + s];
        }
        __syncthreads();
    }
    if ((t & (C - 1)) == 0) {
        int hd = t / C;
        out_s[(size_t)node * (HC / C) + hd] = ss[t];
        out_d[(size_t)node * (HC / C) + hd] = sd[t];
    }
}

// ---------------------------------------------------------------------------
__device__ __forceinline__ float leaky(float x) { return x > 0.f ? x : NEG_SLOPE * x; }

// float atomic max via signed/unsigned int bit trick (init to -inf)
__device__ __forceinline__ void atomicMaxF(float* addr, float val) {
    if (val >= 0.f)
        atomicMax((int*)addr, __float_as_int(val));
    else
        atomicMin((unsigned int*)addr, __float_as_uint(val));
}

__global__ void init_f32(float* __restrict__ p, float val, size_t n) {
    size_t i = (size_t)blockIdx.x * blockDim.x + threadIdx.x;
    if (i < n) p[i] = val;
}

// pass 1: segment max over destination nodes
__global__ void edge_max(const int* __restrict__ src, const int* __restrict__ dst,
                         const float* __restrict__ as, const float* __restrict__ ad,
                         float* __restrict__ m, int ET, int E, int H) {
    int tid = blockIdx.x * blockDim.x + threadIdx.x;
    if (tid >= ET * H) return;
    int e = tid / H, hd = tid - e * H;
    int s, d;
    if (e < E) { s = src[e]; d = dst[e]; } else { s = e - E; d = s; }
    float val = leaky(as[s * H + hd] + ad[d * H + hd]);
    atomicMaxF(&m[d * H + hd], val);
}

// pass 2: segment sum of exp(e - m[dst])
__global__ void edge_sum(const int* __restrict__ src, const int* __restrict__ dst,
                         const float* __restrict__ as, const float* __restrict__ ad,
                         const float* __restrict__ m, float* __restrict__ den,
                         int ET, int E, int H) {
    int tid = blockIdx.x * blockDim.x + threadIdx.x;
    if (tid >= ET * H) return;
    int e = tid / H, hd = tid - e * H;
    int s, d;
    if (e < E) { s = src[e]; d = dst[e]; } else { s = e - E; d = s; }
    float val = leaky(as[s * H + hd] + ad[d * H + hd]);
    float p = expf(val - m[d * H + hd]);
    atomicAdd(&den[d * H + hd], p);
}

// pass 3: out[dst] += h[src] * alpha  (one block per edge, HC threads, coalesced)
__global__ void edge_agg(const int* __restrict__ src, const int* __restrict__ dst,
                         const float* __restrict__ as, const float* __restrict__ ad,
                         const float* __restrict__ m, const float* __restrict__ den,
                         const float* __restrict__ h, float* __restrict__ out,
                         int E, int H, int C) {
    int e = blockIdx.x;
    int t = threadIdx.x;
    int HC = H * C;
    int s, d;
    if (e < E) { s = src[e]; d = dst[e]; } else { s = e - E; d = s; }
    int hd = t / C;
    float val = leaky(as[s * H + hd] + ad[d * H + hd]);
    float p = expf(val - m[d * H + hd]);
    float alpha = p / (den[d * H + hd] + 1e-16f);
    atomicAdd(&out[(size_t)d * HC + t], h[(size_t)s * HC + t] * alpha);
}

// h = elu(agg + bias), in place
__global__ void bias_elu(float* __restrict__ a, const float* __restrict__ bias,
                         size_t n, int HC) {
    size_t i = (size_t)blockIdx.x * blockDim.x + threadIdx.x;
    if (i >= n) return;
    float v = a[i] + bias[i % HC];
    a[i] = v > 0.f ? v : (expf(v) - 1.f);
}

// out += bias, in place
__global__ void add_bias(float* __restrict__ a, const float* __restrict__ bias,
                         size_t n, int HC) {
    size_t i = (size_t)blockIdx.x * blockDim.x + threadIdx.x;
    if (i >= n) return;
    a[i] += bias[i % HC];
}

// ---------------------------------------------------------------------------
extern "C" void kernel_launch(void* const* d_in, const int* in_sizes, int n_in,
                              void* d_out, int out_size, void* d_ws, size_t ws_size,
                              hipStream_t stream) {
    const float* x    = (const float*)d_in[0];
    const int*   ei   = (const int*)d_in[1];
    const float* W1   = (const float*)d_in[2];
    const float* aS1  = (const float*)d_in[3];
    const float* aD1  = (const float*)d_in[4];
    const float* b1   = (const float*)d_in[5];
    const float* W2   = (const float*)d_in[6];
    const float* aS2  = (const float*)d_in[7];
    const float* aD2  = (const float*)d_in[8];
    const float* b2   = (const float*)d_in[9];
    float* out = (float*)d_out;

    const int N = NNODES, E = NEDGES, ET = NEDGES + NNODES;
    const int* srcA = ei;         // edge_index[0]
    const int* dstA = ei + E;     // edge_index[1]

    // workspace carve (floats)
    float* w    = (float*)d_ws;
    float* h1   = w;  w += (size_t)N * 256;   // x @ W1
    float* agg1 = w;  w += (size_t)N * 256;   // layer-1 aggregation / elu output
    float* h2   = w;  w += (size_t)N * 64;    // agg1 @ W2
    float* as1  = w;  w += (size_t)N * 4;
    float* ad1  = w;  w += (size_t)N * 4;
    float* m1   = w;  w += (size_t)N * 4;
    float* den1 = w;  w += (size_t)N * 4;
    float* as2  = w;  w += (size_t)N;
    float* ad2  = w;  w += (size_t)N;
    float* m2   = w;  w += (size_t)N;
    float* den2 = w;  w += (size_t)N;

    auto blks = [](size_t n) { return (unsigned)((n + 255) / 256); };
    const float NEG_INF = -INFINITY;

    // ---------------- Layer 1 ----------------
    // h1 = x @ W1   (M=50000, K=128, Nc=256) -> 3125*16 tiles / 8 waves per block
    gemm_wmma_f32<<<(3125 * 16 + 7) / 8, 256, 0, stream>>>(x, W1, h1, N, 128, 256);
    alpha_kernel<256, 64><<<N, 256, 0, stream>>>(h1, aS1, aD1, as1, ad1);

    init_f32<<<blks((size_t)N * 4), 256, 0, stream>>>(m1, NEG_INF, (size_t)N * 4);
    init_f32<<<blks((size_t)N * 4), 256, 0, stream>>>(den1, 0.f, (size_t)N * 4);
    init_f32<<<blks((size_t)N * 256), 256, 0, stream>>>(agg1, 0.f, (size_t)N * 256);

    edge_max<<<blks((size_t)ET * 4), 256, 0, stream>>>(srcA, dstA, as1, ad1, m1, ET, E, 4);
    edge_sum<<<blks((size_t)ET * 4), 256, 0, stream>>>(srcA, dstA, as1, ad1, m1, den1, ET, E, 4);
    edge_agg<<<ET, 256, 0, stream>>>(srcA, dstA, as1, ad1, m1, den1, h1, agg1, E, 4, 64);

    bias_elu<<<blks((size_t)N * 256), 256, 0, stream>>>(agg1, b1, (size_t)N * 256, 256);

    // ---------------- Layer 2 ----------------
    // h2 = agg1 @ W2  (M=50000, K=256, Nc=64) -> 3125*4 tiles
    gemm_wmma_f32<<<(3125 * 4 + 7) / 8, 256, 0, stream>>>(agg1, W2, h2, N, 256, 64);
    alpha_kernel<64, 64><<<N, 64, 0, stream>>>(h2, aS2, aD2, as2, ad2);

    init_f32<<<blks((size_t)N), 256, 0, stream>>>(m2, NEG_INF, (size_t)N);
    init_f32<<<blks((size_t)N), 256, 0, stream>>>(den2, 0.f, (size_t)N);
    init_f32<<<blks((size_t)N * 64), 256, 0, stream>>>(out, 0.f, (size_t)N * 64);

    edge_max<<<blks((size_t)ET), 256, 0, stream>>>(srcA, dstA, as2, ad2, m2, ET, E, 1);
    edge_sum<<<blks((size_t)ET), 256, 0, stream>>>(srcA, dstA, as2, ad2, m2, den2, ET, E, 1);
    edge_agg<<<ET, 64, 0, stream>>>(srcA, dstA, as2, ad2, m2, den2, h2, out, E, 1, 64);

    add_bias<<<blks((size_t)N * 64), 256, 0, stream>>>(out, b2, (size_t)N * 64, 64);
}